// GAT_12232066859615
// MI455X (gfx1250) — compile-verified
//
#include <hip/hip_runtime.h>
#include <math.h>

// ---------------------------------------------------------------------------
// GAT (2-layer, PyG GATConv semantics) for MI455X / gfx1250.
// GEMMs use V_WMMA_F32_16X16X4_F32 (fp32 in/out, matches reference numerics).
// Column counts are template params so the inner K-loop is branch-free with
// compile-time address math (no per-load EXEC masking / v_mul_u64).
// Edge phase: segment-softmax via ordered-bits atomicMax + f32 atomicAdd;
// whole working set (~90MB) is L2-resident (192MB), so atomics stay on-chip.
// ---------------------------------------------------------------------------

typedef __attribute__((ext_vector_type(2))) float v2f;
typedef __attribute__((ext_vector_type(8))) float v8f;

#define N_NODES 50000
#define N_EDGES 800000
#define FEATS   128
#define HIDDEN  32
#define HEADS   4
#define CLASSES 40
#define NEG_SLOPE 0.2f

// ---- workspace layout (in floats) ----
#define OFF_H1    0ull          // N*128
#define OFF_OUT1  6400000ull    // N*128 (becomes relu'd layer-2 input)
#define OFF_E1    12800000ull   // E*4
#define OFF_AS1   16000000ull   // N*4
#define OFF_AD1   16200000ull   // N*4
#define OFF_M1    16400000ull   // N*4 (uint keys)
#define OFF_D1    16600000ull   // N*4
#define WS_FLOATS 16800000ull   // 67.2 MB total
// layer-2 buffers alias the retired h1 region [0, 6.4M)
#define OFF_H2    0ull          // N*40
#define OFF_OUT2  2000000ull    // N*40
#define OFF_E2    4000000ull    // E
#define OFF_AS2   4800000ull    // N
#define OFF_AD2   4850000ull    // N
#define OFF_M2    4900000ull    // N (uint keys)
#define OFF_D2    4950000ull    // N

// Order-preserving float -> uint mapping (for atomicMax-based segment max).
__device__ __forceinline__ unsigned f2key(float f) {
    unsigned b = __float_as_uint(f);
    return (b & 0x80000000u) ? ~b : (b | 0x80000000u);
}
__device__ __forceinline__ float key2f(unsigned k) {
    return (k & 0x80000000u) ? __uint_as_float(k & 0x7FFFFFFFu)
                             : __uint_as_float(~k);
}

// ---------------------------------------------------------------------------
// WMMA GEMM: C[nrows x NCOLS] = A[nrows x 128] * B[128 x NCOLS], fp32.
// Block tile: 16 rows x (NWAVES*16) cols.
//  - A tile staged in LDS via float4 (padded stride 132 -> the 16 concurrent
//    row-readers in the K-loop hit distinct banks; 16B alignment preserved).
//  - If NCOLS == NWAVES*16: B read straight from global (L2-resident) with
//    compile-time strides -> branch-free, immediate-offset loads.
//  - Else (ragged, e.g. 40 -> 48): B staged zero-padded in LDS once.
// Per-lane fragment layout per CDNA5 ISA 7.12.2:
//   A 16x4 : lanes 0-15 row M=lane {K,K+1}; lanes 16-31 row M=lane-16 {K+2,K+3}
//   B 4x16 : vgpr0 = row K (lanes 0-15) / row K+2 (lanes 16-31); vgpr1 = K+1/K+3
//   C 16x16: vgpr r = row r (lanes 0-15) / row r+8 (lanes 16-31), N = lane&15
// ---------------------------------------------------------------------------
template <int NWAVES, int NCOLS>
__global__ __launch_bounds__(NWAVES * 32)
void gat_gemm_wmma(const float* __restrict__ A, const float* __restrict__ B,
                   float* __restrict__ C) {
    constexpr int NCP     = NWAVES * 16;       // padded column count
    constexpr bool STAGE_B = (NCOLS < NCP);    // ragged edge -> pad via LDS
    constexpr int ASTRIDE = 132;

    __shared__ float As[16 * ASTRIDE];
    __shared__ float Bs[STAGE_B ? FEATS * NCP : 4];

    const int row0 = blockIdx.x * 16;
    const int tid  = threadIdx.x;

    // Stage A tile: 16x128 floats as float4 (512 vec4 loads across the block).
    for (int i = tid; i < 16 * FEATS / 4; i += NWAVES * 32) {
        const int r = i >> 5;              // 32 float4 per row
        const int c = (i & 31) << 2;
        const float4 v = *(const float4*)(A + (size_t)(row0 + r) * FEATS + c);
        *(float4*)(&As[r * ASTRIDE + c]) = v;
    }
    if constexpr (STAGE_B) {
        // Stage B zero-padded to NCP columns (pays the ragged guard once).
        for (int i = tid; i < FEATS * NCP; i += NWAVES * 32) {
            const int r = i / NCP, c = i % NCP;
            Bs[i] = (c < NCOLS) ? B[r * NCOLS + c] : 0.0f;
        }
    }
    __syncthreads();

    const int wave = tid >> 5;
    const int lane = tid & 31;
    const int col0 = wave * 16;
    const int m    = lane & 15;
    const int koff = (lane >> 4) << 1;     // 0 for lanes 0-15, 2 for 16-31
    const int bcol = col0 + m;

    v8f acc = {};
#pragma unroll
    for (int k0 = 0; k0 < FEATS; k0 += 4) {
        v2f a, b;
        a.x = As[m * ASTRIDE + k0 + koff];
        a.y = As[m * ASTRIDE + k0 + koff + 1];
        if constexpr (STAGE_B) {
            b.x = Bs[(k0 + koff) * NCP + bcol];
            b.y = Bs[(k0 + koff + 1) * NCP + bcol];
        } else {
            b.x = B[(k0 + koff) * NCOLS + bcol];
            b.y = B[(k0 + koff + 1) * NCOLS + bcol];
        }
        acc = __builtin_amdgcn_wmma_f32_16x16x4_f32(
            /*neg_a=*/false, a, /*neg_b=*/false, b,
            /*c_mod=*/(short)0, acc, /*reuse_a=*/false, /*reuse_b=*/false);
    }

    if (bcol < NCOLS) {                    // compile-time true when not ragged
        const int orow = row0 + ((lane >> 4) << 3);
#pragma unroll
        for (int r = 0; r < 8; ++r)
            C[(size_t)(orow + r) * NCOLS + bcol] = acc[r];
    }
}

// ---- attention coefficients per node -------------------------------------
__global__ void gat_alphas1(const float* __restrict__ h,
                            const float* __restrict__ asrc,
                            const float* __restrict__ adst,
                            float* __restrict__ s_out, float* __restrict__ d_out) {
    const unsigned i = blockIdx.x * blockDim.x + threadIdx.x;
    if (i >= N_NODES * HEADS) return;
    const unsigned n = i >> 2, hh = i & 3;
    const float* hp = h + (size_t)n * FEATS + hh * HIDDEN;
    const float* sp = asrc + hh * HIDDEN;
    const float* dp = adst + hh * HIDDEN;
    float s = 0.f, d = 0.f;
#pragma unroll
    for (int c = 0; c < HIDDEN; ++c) { s += hp[c] * sp[c]; d += hp[c] * dp[c]; }
    s_out[i] = s; d_out[i] = d;
}

__global__ void gat_alphas2(const float* __restrict__ h,
                            const float* __restrict__ asrc,
                            const float* __restrict__ adst,
                            float* __restrict__ s_out, float* __restrict__ d_out) {
    const unsigned n = blockIdx.x * blockDim.x + threadIdx.x;
    if (n >= N_NODES) return;
    const float* hp = h + (size_t)n * CLASSES;
    float s = 0.f, d = 0.f;
#pragma unroll
    for (int c = 0; c < CLASSES; ++c) { s += hp[c] * asrc[c]; d += hp[c] * adst[c]; }
    s_out[n] = s; d_out[n] = d;
}

// ---- edge pass 1: leaky-relu logits + segment max (ordered-bits atomics) --
template <int H>
__global__ void gat_edge_logits(const int* __restrict__ ei,
                                const float* __restrict__ as,
                                const float* __restrict__ ad,
                                float* __restrict__ ebuf,
                                unsigned* __restrict__ mkey) {
    const unsigned i = blockIdx.x * blockDim.x + threadIdx.x;
    if (i >= (unsigned)N_EDGES * H) return;
    const unsigned e = i / H, hh = i % H;
    const int src = ei[e], dst = ei[N_EDGES + e];
    float v = as[(size_t)src * H + hh] + ad[(size_t)dst * H + hh];
    v = v > 0.f ? v : NEG_SLOPE * v;
    ebuf[i] = v;
    atomicMax(&mkey[(size_t)dst * H + hh], f2key(v));
}

// ---- edge pass 2: exp(e - max) + segment sum ------------------------------
template <int H>
__global__ void gat_edge_expsum(const int* __restrict__ ei,
                                float* __restrict__ ebuf,
                                const unsigned* __restrict__ mkey,
                                float* __restrict__ denom) {
    const unsigned i = blockIdx.x * blockDim.x + threadIdx.x;
    if (i >= (unsigned)N_EDGES * H) return;
    const unsigned e = i / H, hh = i % H;
    const int dst = ei[N_EDGES + e];
    const float mm = key2f(mkey[(size_t)dst * H + hh]);
    const float p = expf(ebuf[i] - mm);
    ebuf[i] = p;
    atomicAdd(&denom[(size_t)dst * H + hh], p);
}

// ---- edge pass 3: weighted message scatter (1 thread per edge-element) ----
template <int H, int C>
__global__ void gat_edge_scatter(const int* __restrict__ ei,
                                 const float* __restrict__ h,
                                 const float* __restrict__ ebuf,
                                 const float* __restrict__ denom,
                                 float* __restrict__ out) {
    constexpr int F = H * C;
    const unsigned i = blockIdx.x * blockDim.x + threadIdx.x;
    if (i >= (unsigned)N_EDGES * F) return;
    const unsigned e = i / F, f = i % F;
    const unsigned hh = f / C;
    const int src = ei[e], dst = ei[N_EDGES + e];
    const float coef = ebuf[(size_t)e * H + hh] /
                       (denom[(size_t)dst * H + hh] + 1e-16f);
    atomicAdd(&out[(size_t)dst * F + f], h[(size_t)src * F + f] * coef);
}

__global__ void gat_bias_relu(float* __restrict__ out, const float* __restrict__ bias) {
    const unsigned i = blockIdx.x * blockDim.x + threadIdx.x;
    if (i >= (unsigned)N_NODES * FEATS) return;
    const float v = out[i] + bias[i & 127];
    out[i] = v > 0.f ? v : 0.f;
}

__global__ void gat_bias_logsoftmax(const float* __restrict__ acc,
                                    const float* __restrict__ bias,
                                    float* __restrict__ out) {
    const unsigned n = blockIdx.x * blockDim.x + threadIdx.x;
    if (n >= N_NODES) return;
    float v[CLASSES];
    float mx = -INFINITY;
#pragma unroll
    for (int c = 0; c < CLASSES; ++c) {
        v[c] = acc[(size_t)n * CLASSES + c] + bias[c];
        mx = fmaxf(mx, v[c]);
    }
    float s = 0.f;
#pragma unroll
    for (int c = 0; c < CLASSES; ++c) s += expf(v[c] - mx);
    const float lse = logf(s);
#pragma unroll
    for (int c = 0; c < CLASSES; ++c)
        out[(size_t)n * CLASSES + c] = v[c] - mx - lse;
}

extern "C" void kernel_launch(void* const* d_in, const int* in_sizes, int n_in,
                              void* d_out, int out_size, void* d_ws, size_t ws_size,
                              hipStream_t stream) {
    const float* x   = (const float*)d_in[0];
    const float* W1  = (const float*)d_in[1];
    const float* as1 = (const float*)d_in[2];
    const float* ad1 = (const float*)d_in[3];
    const float* b1  = (const float*)d_in[4];
    const float* W2  = (const float*)d_in[5];
    const float* as2 = (const float*)d_in[6];
    const float* ad2 = (const float*)d_in[7];
    const float* b2  = (const float*)d_in[8];
    const int*   ei  = (const int*)d_in[9];

    float* ws  = (float*)d_ws;
    float* out = (float*)d_out;

    float*    h1   = ws + OFF_H1;
    float*    out1 = ws + OFF_OUT1;
    float*    e1   = ws + OFF_E1;
    float*    aS1  = ws + OFF_AS1;
    float*    aD1  = ws + OFF_AD1;
    unsigned* m1   = (unsigned*)(ws + OFF_M1);
    float*    dn1  = ws + OFF_D1;

    // ---- layer 1 ----
    hipMemsetAsync(out1, 0, sizeof(float) * (size_t)N_NODES * FEATS, stream);
    hipMemsetAsync(ws + OFF_M1, 0, sizeof(float) * 2 * (size_t)N_NODES * HEADS, stream);

    gat_gemm_wmma<8, 128><<<N_NODES / 16, 256, 0, stream>>>(x, W1, h1);
    gat_alphas1<<<(N_NODES * HEADS + 255) / 256, 256, 0, stream>>>(h1, as1, ad1, aS1, aD1);
    gat_edge_logits<HEADS><<<(N_EDGES * HEADS + 255) / 256, 256, 0, stream>>>(ei, aS1, aD1, e1, m1);
    gat_edge_expsum<HEADS><<<(N_EDGES * HEADS + 255) / 256, 256, 0, stream>>>(ei, e1, m1, dn1);
    gat_edge_scatter<HEADS, HIDDEN>
        <<<(unsigned)(((size_t)N_EDGES * FEATS + 255) / 256), 256, 0, stream>>>(ei, h1, e1, dn1, out1);
    gat_bias_relu<<<(N_NODES * FEATS + 255) / 256, 256, 0, stream>>>(out1, b1);

    // ---- layer 2 (buffers alias the retired h1 region) ----
    float*    h2   = ws + OFF_H2;
    float*    out2 = ws + OFF_OUT2;
    float*    e2   = ws + OFF_E2;
    float*    aS2  = ws + OFF_AS2;
    float*    aD2  = ws + OFF_AD2;
    unsigned* m2   = (unsigned*)(ws + OFF_M2);
    float*    dn2  = ws + OFF_D2;

    hipMemsetAsync(out2, 0, sizeof(float) * (size_t)N_NODES * CLASSES, stream);
    hipMemsetAsync(ws + OFF_M2, 0, sizeof(float) * 2 * (size_t)N_NODES, stream);

    gat_gemm_wmma<3, CLASSES><<<N_NODES / 16, 96, 0, stream>>>(out1, W2, h2);
    gat_alphas2<<<(N_NODES + 255) / 256, 256, 0, stream>>>(h2, as2, ad2, aS2, aD2);
    gat_edge_logits<1><<<(N_EDGES + 255) / 256, 256, 0, stream>>>(ei, aS2, aD2, e2, m2);
    gat_edge_expsum<1><<<(N_EDGES + 255) / 256, 256, 0, stream>>>(ei, e2, m2, dn2);
    gat_edge_scatter<1, CLASSES>
        <<<(unsigned)(((size_t)N_EDGES * CLASSES + 255) / 256), 256, 0, stream>>>(ei, h2, e2, dn2, out2);
    gat_bias_logsoftmax<<<(N_NODES + 255) / 256, 256, 0, stream>>>(out2, b2, out);
}